// CustomGRUModel_78280073937754
// MI455X (gfx1250) — compile-verified
//
#include <hip/hip_runtime.h>
#include <math.h>

// GRU: B=256, T=512, D=256, H=512, O=1.
// Phase 1: gx[t*B+b][k] = sum_d x[b][t][d]*W[d][k]      (WMMA f32 GEMM, parallel)
// Phase 2: persistent scan kernel, 64 blocks, software grid barrier, 2 GEMM
//          stages per time step (reference uses (r*h)@Un, so r must be
//          materialized before the n-gate GEMM).
// Phase 3: out = h_last @ Wd + bd.
// All matrix math uses V_WMMA_F32_16X16X4_F32 to preserve fp32 semantics.

#define Bb   256
#define Tt   512
#define Dd   256
#define Hh   512
#define H3   1536
#define SCAN_BLOCKS 64

typedef __attribute__((ext_vector_type(2))) float v2f;
typedef __attribute__((ext_vector_type(8))) float v8f;

__device__ __forceinline__ v8f wmma_f32(v2f a, v2f b, v8f c) {
  // 8 args: (neg_a, A, neg_b, B, c_mod, C, reuse_a, reuse_b)
  return __builtin_amdgcn_wmma_f32_16x16x4_f32(false, a, false, b, (short)0, c,
                                               false, false);
}

__device__ __forceinline__ v8f v8zero() {
  v8f z = {0.f, 0.f, 0.f, 0.f, 0.f, 0.f, 0.f, 0.f};
  return z;
}

// ---------------------------------------------------------------------------
// Kernel 1: gx = x @ W.  M=131072 (m = t*256+b), K=256, N=1536.
// Block tile 128x128, 8 waves (2x4), wave tile 64x32 (4x2 WMMA tiles).
// ---------------------------------------------------------------------------
__global__ __launch_bounds__(256) void gemm_xw(const float* __restrict__ x,
                                               const float* __restrict__ W,
                                               float* __restrict__ gx) {
  __shared__ float xs[128 * 36];   // A tile, padded stride 36 (conflict-free)
  __shared__ float wsm[32 * 136];  // B tile, padded stride 136

  const int tid  = threadIdx.x;
  const int lane = tid & 31;
  const int wave = tid >> 5;
  const int wm   = wave >> 2;          // 0..1
  const int wn   = wave & 3;           // 0..3
  const int lcol = lane & 15;
  const int lhalf = lane >> 4;
  const int klo  = lhalf * 2;

  const int tn = blockIdx.x;           // 0..11  (N tiles of 128)
  const int tm = blockIdx.y;           // 0..1023 (M tiles of 128)
  const int t  = tm >> 1;              // each 128-row tile has constant t
  const int b0 = (tm & 1) * 128;

  v8f acc[4][2];
#pragma unroll
  for (int mi = 0; mi < 4; ++mi)
#pragma unroll
    for (int ni = 0; ni < 2; ++ni) acc[mi][ni] = v8zero();

  for (int k0 = 0; k0 < Dd; k0 += 32) {
#pragma unroll
    for (int i = 0; i < 4; ++i) {      // x tile: 128 rows x 32 k
      int idx = tid + i * 256;
      int r = idx >> 3, q = idx & 7;
      float4 v = *(const float4*)(x + (size_t)(b0 + r) * (Tt * Dd) +
                                  (size_t)t * Dd + k0 + q * 4);
      *(float4*)(xs + r * 36 + q * 4) = v;
    }
#pragma unroll
    for (int i = 0; i < 4; ++i) {      // W tile: 32 k x 128 n
      int idx = tid + i * 256;
      int r = idx >> 5, q = idx & 31;
      float4 v = *(const float4*)(W + (size_t)(k0 + r) * H3 + tn * 128 + q * 4);
      *(float4*)(wsm + r * 136 + q * 4) = v;
    }
    __syncthreads();
#pragma unroll
    for (int kk = 0; kk < 32; kk += 4) {
      v2f a[4], bf[2];
#pragma unroll
      for (int mi = 0; mi < 4; ++mi) {
        int r = (wm * 64 + mi * 16 + lcol) * 36;
        a[mi].x = xs[r + kk + klo];
        a[mi].y = xs[r + kk + klo + 1];
      }
#pragma unroll
      for (int ni = 0; ni < 2; ++ni) {
        int c = wn * 32 + ni * 16 + lcol;
        bf[ni].x = wsm[(kk + klo) * 136 + c];
        bf[ni].y = wsm[(kk + klo + 1) * 136 + c];
      }
#pragma unroll
      for (int mi = 0; mi < 4; ++mi)
#pragma unroll
        for (int ni = 0; ni < 2; ++ni)
          acc[mi][ni] = wmma_f32(a[mi], bf[ni], acc[mi][ni]);
    }
    __syncthreads();
  }

  const size_t mBase = (size_t)tm * 128 + wm * 64;
#pragma unroll
  for (int mi = 0; mi < 4; ++mi) {
#pragma unroll
    for (int ni = 0; ni < 2; ++ni) {
      int colG = tn * 128 + wn * 32 + ni * 16 + lcol;
#pragma unroll
      for (int i = 0; i < 8; ++i) {
        size_t row = mBase + mi * 16 + lhalf * 8 + i;  // C: M=i / M=i+8
        gx[row * H3 + colG] = acc[mi][ni][i];
      }
    }
  }
}

// ---------------------------------------------------------------------------
// Grid barrier: monotonic counter, agent scope, fences flush/invalidate caches
// ---------------------------------------------------------------------------
__device__ __forceinline__ void grid_barrier(unsigned* cnt, unsigned target) {
  __threadfence();   // release: make our global writes visible at device scope
  __syncthreads();
  if (threadIdx.x == 0) {
    __hip_atomic_fetch_add(cnt, 1u, __ATOMIC_ACQ_REL, __HIP_MEMORY_SCOPE_AGENT);
    while (__hip_atomic_load(cnt, __ATOMIC_RELAXED, __HIP_MEMORY_SCOPE_AGENT) <
           target) {
      __builtin_amdgcn_s_sleep(1);
    }
  }
  __syncthreads();
  __threadfence();   // acquire: invalidate stale lines before reading peers'
}

// ---------------------------------------------------------------------------
// Kernel 2: persistent GRU scan. 64 blocks x 256 threads (8 waves).
// Stage A: ZR = sigmoid(gx_t[:, :1024] + h@U[:, :1024] + b[:1024])
//          -> store z, and rh = r*h.  Tile grid 8(M=32) x 8(N=128).
// Stage B: n = tanh(gx_t[:,1024:] + rh@U[:,1024:] + b[1024:])
//          h = z*h + (1-z)*n.         Tile grid 8(M=32) x 8(N=64).
// ---------------------------------------------------------------------------
__global__ __launch_bounds__(256) void gru_scan(const float* __restrict__ gx,
                                                const float* __restrict__ U,
                                                const float* __restrict__ bias,
                                                float* __restrict__ h,
                                                float* __restrict__ zb,
                                                float* __restrict__ rh,
                                                unsigned* __restrict__ bar) {
  __shared__ float sA[32 * 36];
  __shared__ float sB[32 * 136];

  const int tid  = threadIdx.x;
  const int lane = tid & 31;
  const int wave = tid >> 5;
  const int wm   = wave >> 2;
  const int wn   = wave & 3;
  const int lcol = lane & 15;
  const int lhalf = lane >> 4;
  const int klo  = lhalf * 2;
  const int bm   = blockIdx.x >> 3;   // 0..7  (rows of 32)
  const int bn   = blockIdx.x & 7;    // 0..7

  unsigned target = 0;

  for (int t = 0; t < Tt; ++t) {
    const float* gxt = gx + (size_t)t * (Bb * H3);

    // ---------------- Stage A ----------------
    {
      v8f acc[2] = {v8zero(), v8zero()};
      const int rowBlk = bm * 32;
      const int colBlk = bn * 128;       // column in [0,1024)
      for (int k0 = 0; k0 < Hh; k0 += 32) {
        {  // h tile 32x32
          int r = tid >> 3, q = tid & 7;
          *(float4*)(sA + r * 36 + q * 4) =
              *(const float4*)(h + (size_t)(rowBlk + r) * Hh + k0 + q * 4);
        }
#pragma unroll
        for (int i = 0; i < 4; ++i) {  // U tile 32x128
          int idx = tid + i * 256;
          int r = idx >> 5, q = idx & 31;
          *(float4*)(sB + r * 136 + q * 4) =
              *(const float4*)(U + (size_t)(k0 + r) * H3 + colBlk + q * 4);
        }
        __syncthreads();
#pragma unroll
        for (int kk = 0; kk < 32; kk += 4) {
          v2f a;
          int ar = (wm * 16 + lcol) * 36;
          a.x = sA[ar + kk + klo];
          a.y = sA[ar + kk + klo + 1];
#pragma unroll
          for (int ni = 0; ni < 2; ++ni) {
            v2f bb;
            int c = wn * 32 + ni * 16 + lcol;
            bb.x = sB[(kk + klo) * 136 + c];
            bb.y = sB[(kk + klo + 1) * 136 + c];
            acc[ni] = wmma_f32(a, bb, acc[ni]);
          }
        }
        __syncthreads();
      }
#pragma unroll
      for (int ni = 0; ni < 2; ++ni) {
        int colG = colBlk + wn * 32 + ni * 16 + lcol;  // [0,1024)
        float bv = bias[colG];
#pragma unroll
        for (int i = 0; i < 8; ++i) {
          int row = rowBlk + wm * 16 + lhalf * 8 + i;
          float pre = acc[ni][i] + gxt[(size_t)row * H3 + colG] + bv;
          float s = 1.f / (1.f + __expf(-pre));
          if (colG < Hh) {
            zb[(size_t)row * Hh + colG] = s;          // z gate
          } else {
            int jj = colG - Hh;                        // r gate -> r*h
            rh[(size_t)row * Hh + jj] = s * h[(size_t)row * Hh + jj];
          }
        }
      }
    }
    target += SCAN_BLOCKS;
    grid_barrier(bar, target);

    // ---------------- Stage B ----------------
    {
      v8f acc = v8zero();
      const int rowBlk = bm * 32;
      const int colBlk = bn * 64;        // column in [0,512)
      for (int k0 = 0; k0 < Hh; k0 += 32) {
        {  // rh tile 32x32
          int r = tid >> 3, q = tid & 7;
          *(float4*)(sA + r * 36 + q * 4) =
              *(const float4*)(rh + (size_t)(rowBlk + r) * Hh + k0 + q * 4);
        }
#pragma unroll
        for (int i = 0; i < 2; ++i) {  // Un tile 32x64 (stride 72)
          int idx = tid + i * 256;
          int r = idx >> 4, q = idx & 15;
          *(float4*)(sB + r * 72 + q * 4) = *(const float4*)(
              U + (size_t)(k0 + r) * H3 + 2 * Hh + colBlk + q * 4);
        }
        __syncthreads();
#pragma unroll
        for (int kk = 0; kk < 32; kk += 4) {
          v2f a;
          int ar = (wm * 16 + lcol) * 36;
          a.x = sA[ar + kk + klo];
          a.y = sA[ar + kk + klo + 1];
          v2f bb;
          int c = wn * 16 + lcol;
          bb.x = sB[(kk + klo) * 72 + c];
          bb.y = sB[(kk + klo + 1) * 72 + c];
          acc = wmma_f32(a, bb, acc);
        }
        __syncthreads();
      }
      int colG = colBlk + wn * 16 + lcol;  // [0,512)
      float bv = bias[2 * Hh + colG];
#pragma unroll
      for (int i = 0; i < 8; ++i) {
        int row = rowBlk + wm * 16 + lhalf * 8 + i;
        float pre = acc[i] + gxt[(size_t)row * H3 + 2 * Hh + colG] + bv;
        float n = tanhf(pre);
        size_t off = (size_t)row * Hh + colG;
        float zv = zb[off];
        float hv = h[off];
        h[off] = zv * hv + (1.f - zv) * n;   // in-place: only (row,colG) owner
      }
    }
    target += SCAN_BLOCKS;
    grid_barrier(bar, target);
  }
}

// ---------------------------------------------------------------------------
// Kernel 3: out = h_last @ Wd + bd   (256x512 * 512x1)
// ---------------------------------------------------------------------------
__global__ __launch_bounds__(256) void head_kernel(const float* __restrict__ h,
                                                   const float* __restrict__ Wd,
                                                   const float* __restrict__ bd,
                                                   float* __restrict__ out) {
  int b = threadIdx.x;
  float s = bd[0];
  for (int j = 0; j < Hh; ++j) s += h[(size_t)b * Hh + j] * Wd[j];
  out[b] = s;
}

// ---------------------------------------------------------------------------
extern "C" void kernel_launch(void* const* d_in, const int* in_sizes, int n_in,
                              void* d_out, int out_size, void* d_ws,
                              size_t ws_size, hipStream_t stream) {
  const float* x  = (const float*)d_in[0];
  const float* W  = (const float*)d_in[1];
  const float* U  = (const float*)d_in[2];
  const float* b  = (const float*)d_in[3];
  const float* Wd = (const float*)d_in[4];
  const float* bd = (const float*)d_in[5];
  float* out = (float*)d_out;

  // Workspace layout (needs ~807 MB):
  //   gx: T*B*3H fp32   (805,306,368 B)
  //   h : B*H fp32      (524,288 B)   <- zeroed each launch
  //   bar: 256 B                       <- zeroed each launch
  //   zb: B*H fp32, rh: B*H fp32
  const size_t GX_BYTES = (size_t)Tt * Bb * H3 * sizeof(float);
  const size_t HB       = (size_t)Bb * Hh * sizeof(float);
  char* ws = (char*)d_ws;
  float*    gx  = (float*)ws;
  float*    h   = (float*)(ws + GX_BYTES);
  unsigned* bar = (unsigned*)(ws + GX_BYTES + HB);
  float*    zb  = (float*)(ws + GX_BYTES + HB + 256);
  float*    rh  = (float*)(ws + GX_BYTES + HB + 256 + HB);

  hipMemsetAsync(h, 0, HB + 256, stream);  // h0 = 0, barrier counter = 0

  gemm_xw<<<dim3(H3 / 128, (Bb * Tt) / 128), 256, 0, stream>>>(x, W, gx);
  gru_scan<<<SCAN_BLOCKS, 256, 0, stream>>>(gx, U, b, h, zb, rh, bar);
  head_kernel<<<1, 256, 0, stream>>>(h, Wd, bd, out);
}